// Model_78451872628851
// MI455X (gfx1250) — compile-verified
//
#include <hip/hip_runtime.h>
#include <math.h>

// ---------------------------------------------------------------------------
// Problem constants (reference: B=64, T=512, D=H=256)
// ---------------------------------------------------------------------------
#define B_  64
#define T_  512
#define D_  256
#define H_  256
#define G4_ 1024   // 4*H

typedef float v8f __attribute__((ext_vector_type(8)));
typedef float v2f __attribute__((ext_vector_type(2)));
typedef int   v4i __attribute__((ext_vector_type(4)));

// Optional gfx1250 async global->LDS path (ASYNCcnt-tracked), with safe
// fallback to manual global_load_b128 + ds_store staging.
#if defined(__has_builtin)
#if __has_builtin(__builtin_amdgcn_global_load_async_to_lds_b128) && \
    __has_builtin(__builtin_amdgcn_s_wait_asynccnt)
#define USE_ASYNC_LDS 1
#endif
#endif

#if defined(USE_ASYNC_LDS)
// param0: int4 in addrspace(1) (diagnostic showed "int4 __device__ *")
// param1: int4 in addrspace(3)
typedef __attribute__((address_space(1))) v4i g_v4i;
typedef __attribute__((address_space(3))) v4i l_v4i;
#endif

// fp32 WMMA: D(16x16,f32) = A(16x4,f32) * B(4x16,f32) + C
__device__ __forceinline__ v8f wmma_f32(v2f a, v2f b, v8f c) {
  return __builtin_amdgcn_wmma_f32_16x16x4_f32(false, a, false, b, (short)0, c,
                                               false, false);
}

__device__ __forceinline__ float sigm(float x) {
  return 1.0f / (1.0f + expf(-x));
}

// ---------------------------------------------------------------------------
// Kernel 0: pack W_ih / W_hh rows into gate-interleaved layout
//   j' = 4*hcol + gate   (gate order i,f,g,o)   <-   j = gate*H + hcol
// ---------------------------------------------------------------------------
__global__ void pack_kernel(const float* __restrict__ W_ih,
                            const float* __restrict__ W_hh,
                            const float* __restrict__ b_ih,
                            const float* __restrict__ b_hh,
                            float* __restrict__ WpI,
                            float* __restrict__ WpH,
                            float* __restrict__ bp) {
  const int jp   = blockIdx.x;        // 0..1023 packed row
  const int hcol = jp >> 2;
  const int gate = jp & 3;
  const int j    = gate * H_ + hcol;
  const int k    = threadIdx.x;       // 0..255
  WpI[(size_t)jp * D_ + k] = W_ih[(size_t)j * D_ + k];
  WpH[(size_t)jp * H_ + k] = W_hh[(size_t)j * H_ + k];
  if (k == 0) bp[jp] = b_ih[j] + b_hh[j];
}

// ---------------------------------------------------------------------------
// Kernel 1/3: C[m,n] = sum_k A[m,k] * W[n,k] + bias[n]
// Block: 128 threads (4 waves), tile 64(M) x 64(N), K staged in LDS by 32.
// Fragments are double-buffered one k-step ahead so WMMAs overlap ds loads.
// ---------------------------------------------------------------------------
#define KC 32
__global__ __launch_bounds__(128)
void gemm_bias_kernel(const float* __restrict__ A, const float* __restrict__ W,
                      const float* __restrict__ bias, float* __restrict__ C,
                      int K, int N) {
  __shared__ float As[64][KC + 1];   // +1 pad: conflict-free frag reads
  __shared__ float Bs[64][KC + 1];   // Bs[n_local][k] = W[colBase+n][k]

  const int tid  = threadIdx.x;
  const int wav  = tid >> 5;
  const int lane = tid & 31;
  const int half = lane >> 4;
  const int l16  = lane & 15;

  const int rowBase = blockIdx.y * 64;
  const int colBase = blockIdx.x * 64;

  const v8f vzero = {};
  v8f acc[4] = {vzero, vzero, vzero, vzero};

#pragma unroll 1
  for (int k0 = 0; k0 < K; k0 += KC) {
    // stage 64x32 of A and of W (512 16B-chunks / 128 threads)
#pragma unroll
    for (int i = 0; i < 4; ++i) {
      const int idx = tid + 128 * i;   // 0..511
      const int r   = idx >> 3;
      const int s   = idx & 7;
      const float* ga = A + (size_t)(rowBase + r) * K + k0 + s * 4;
      const float* gw = W + (size_t)(colBase + r) * K + k0 + s * 4;
#if defined(USE_ASYNC_LDS)
      __builtin_amdgcn_global_load_async_to_lds_b128(
          (g_v4i*)ga, (l_v4i*)&As[r][s * 4], 0, 0);
      __builtin_amdgcn_global_load_async_to_lds_b128(
          (g_v4i*)gw, (l_v4i*)&Bs[r][s * 4], 0, 0);
#else
      const float4 av = *(const float4*)ga;
      As[r][s * 4 + 0] = av.x; As[r][s * 4 + 1] = av.y;
      As[r][s * 4 + 2] = av.z; As[r][s * 4 + 3] = av.w;
      const float4 wv = *(const float4*)gw;
      Bs[r][s * 4 + 0] = wv.x; Bs[r][s * 4 + 1] = wv.y;
      Bs[r][s * 4 + 2] = wv.z; Bs[r][s * 4 + 3] = wv.w;
#endif
    }
#if defined(USE_ASYNC_LDS)
    __builtin_amdgcn_s_wait_asynccnt(0);
#endif
    __syncthreads();

    // double-buffered fragment pipeline over the 8 k-steps of this chunk
    const int ar = wav * 16 + l16;
    v2f a_c, a_n, b_c[4], b_n[4];
    {
      const int ka = 2 * half;
      a_c.x = As[ar][ka]; a_c.y = As[ar][ka + 1];
#pragma unroll
      for (int nt = 0; nt < 4; ++nt) {
        b_c[nt].x = Bs[nt * 16 + l16][ka];
        b_c[nt].y = Bs[nt * 16 + l16][ka + 1];
      }
    }
#pragma unroll
    for (int kk = 0; kk < KC; kk += 4) {
      const int kn = ((kk + 4) & (KC - 1)) + 2 * half;  // next (wraps, benign)
      a_n.x = As[ar][kn]; a_n.y = As[ar][kn + 1];
#pragma unroll
      for (int nt = 0; nt < 4; ++nt) {
        b_n[nt].x = Bs[nt * 16 + l16][kn];
        b_n[nt].y = Bs[nt * 16 + l16][kn + 1];
      }
#pragma unroll
      for (int nt = 0; nt < 4; ++nt)
        acc[nt] = wmma_f32(a_c, b_c[nt], acc[nt]);
      a_c = a_n;
#pragma unroll
      for (int nt = 0; nt < 4; ++nt) b_c[nt] = b_n[nt];
    }
    __syncthreads();
  }

  // epilogue: bias + store. D layout: row = i + 8*half, col = l16.
#pragma unroll
  for (int nt = 0; nt < 4; ++nt) {
    const int col = colBase + nt * 16 + l16;
    const float bv = bias[col];
#pragma unroll
    for (int i = 0; i < 8; ++i) {
      const int row = rowBase + wav * 16 + i + 8 * half;
      C[(size_t)row * N + col] = acc[nt][i] + bv;
    }
  }
}

// ---------------------------------------------------------------------------
// Kernel 2: a_tj[b,t] = sigmoid( dot(vlin[b,0,:], values[b,t,:]) )
// ---------------------------------------------------------------------------
__global__ __launch_bounds__(256)
void attn_kernel(const float* __restrict__ vlin,
                 const float* __restrict__ values, float* __restrict__ atj) {
  const int b = blockIdx.x;
  __shared__ float v0[D_];
  for (int d = threadIdx.x; d < D_; d += blockDim.x)
    v0[d] = vlin[(size_t)b * T_ * D_ + d];
  __syncthreads();
  for (int t = threadIdx.x; t < T_; t += blockDim.x) {
    const float4* vp = (const float4*)(values + ((size_t)b * T_ + t) * D_);
    float s = 0.f;
#pragma unroll 4
    for (int d = 0; d < D_ / 4; ++d) {
      const float4 v = vp[d];
      s += v0[4 * d + 0] * v.x + v0[4 * d + 1] * v.y +
           v0[4 * d + 2] * v.z + v0[4 * d + 3] * v.w;
    }
    atj[b * T_ + t] = sigm(s);
  }
}

// ---------------------------------------------------------------------------
// Kernel 4: sequential scan. One persistent workgroup, 32 waves.
// h in LDS [64][HP]; recurrent gates via fp32 WMMA from packed W_hh (global,
// cache-resident); c / c0 / cumsum(Deltas) live in registers for all steps.
// ---------------------------------------------------------------------------
#define HP 257          // hbuf row pitch (pad)
#define SP 68           // strip row pitch (16B-aligned, padded)

__global__ __launch_bounds__(1024)
void scan_kernel(const float* __restrict__ Xg,   // [B,T,1024] packed gates(x)
                 const float* __restrict__ WpH,  // [1024,256] packed W_hh
                 const float* __restrict__ atj,  // [B,T]
                 const float* __restrict__ Deltas,  // [B,T,256]
                 float* __restrict__ hn) {          // [B,T,256] output
  extern __shared__ float lds[];
  float* hbuf = lds;                                   // 64 x HP
  const int w    = threadIdx.x >> 5;
  const int lane = threadIdx.x & 31;
  const int half = lane >> 4;
  const int l16  = lane & 15;
  float* strip = lds + 64 * HP + w * 16 * SP;          // wave-private 16 x SP

  int rt[2], cg[2];
#pragma unroll
  for (int j = 0; j < 2; ++j) {
    const int jb = w + 32 * j;
    rt[j] = jb & 3;
    cg[j] = jb >> 2;
  }
  int pr[8], ph[8];
#pragma unroll
  for (int p = 0; p < 8; ++p) {
    pr[p] = (lane >> 2) + 8 * (p & 1);
    ph[p] = (lane & 3) * 4 + (p >> 1);
  }

  float cv[2][8], c0v[2][8], dacc[2][8], hreg[2][8];

  // ---- step t = 0: gates = Xg[:,0,:] (h0 == 0); carry c stays zero ----
#pragma unroll
  for (int j = 0; j < 2; ++j) {
#pragma unroll
    for (int p = 0; p < 8; ++p) {
      const int b    = rt[j] * 16 + pr[p];
      const int hcol = cg[j] * 16 + ph[p];
      const float4 g4 =
          *(const float4*)(Xg + ((size_t)b * T_) * G4_ + 4 * hcol);
      const float gi = sigm(g4.x);
      const float gg = tanhf(g4.z), go = sigm(g4.w);
      const float c0 = gi * gg;          // f*0 + i*tanh(g)
      const float h  = go * tanhf(c0);
      c0v[j][p]  = c0;
      cv[j][p]   = 0.f;                  // faithful quirk: carry stays 0
      dacc[j][p] = Deltas[((size_t)b * T_) * H_ + hcol];
      hbuf[b * HP + hcol] = h;
      hn[((size_t)b * T_) * H_ + hcol] = h;
    }
  }
  __syncthreads();

  // ---- steps t = 1..T-1 ----
#pragma unroll 1
  for (int t = 1; t < T_; ++t) {
#pragma unroll 1
    for (int j = 0; j < 2; ++j) {
      const v8f vzero = {};
      v8f acc[4] = {vzero, vzero, vzero, vzero};
      const float* hrow = hbuf + (rt[j] * 16 + l16) * HP;
      const float* wr[4];
#pragma unroll
      for (int nt = 0; nt < 4; ++nt)
        wr[nt] = WpH + (size_t)(cg[j] * 64 + nt * 16 + l16) * H_;

      // prefetch next timestep's x-gates for this job (global_prefetch_b8)
      if (t + 1 < T_)
        __builtin_prefetch(
            Xg + ((size_t)(rt[j] * 16 + pr[0]) * T_ + t + 1) * G4_ +
                4 * (cg[j] * 16 + ph[0]), 0, 1);

      // double-buffered fragments: a from LDS (DScnt), b from L2 (LOADcnt)
      v2f a_c, a_n, b_c[4], b_n[4];
      {
        const int ka = 2 * half;
        a_c.x = hrow[ka]; a_c.y = hrow[ka + 1];
#pragma unroll
        for (int nt = 0; nt < 4; ++nt) b_c[nt] = *(const v2f*)(wr[nt] + ka);
      }
#pragma unroll 4
      for (int kk = 0; kk < H_; kk += 4) {
        const int kn = ((kk + 4) & (H_ - 1)) + 2 * half;
        a_n.x = hrow[kn]; a_n.y = hrow[kn + 1];
#pragma unroll
        for (int nt = 0; nt < 4; ++nt) b_n[nt] = *(const v2f*)(wr[nt] + kn);
#pragma unroll
        for (int nt = 0; nt < 4; ++nt)
          acc[nt] = wmma_f32(a_c, b_c[nt], acc[nt]);
        a_c = a_n;
#pragma unroll
        for (int nt = 0; nt < 4; ++nt) b_c[nt] = b_n[nt];
      }

      // scatter 16x64 gate strip into wave-private LDS
#pragma unroll
      for (int nt = 0; nt < 4; ++nt)
#pragma unroll
        for (int i = 0; i < 8; ++i)
          strip[(i + 8 * half) * SP + nt * 16 + l16] = acc[nt][i];

      // elementwise: gather (i,f,g,o) as float4, update c in registers
#pragma unroll
      for (int p = 0; p < 8; ++p) {
        const int b    = rt[j] * 16 + pr[p];
        const int hcol = cg[j] * 16 + ph[p];
        const float4 s4 = *(const float4*)(strip + pr[p] * SP + 4 * ph[p]);
        const float4 x4 =
            *(const float4*)(Xg + ((size_t)b * T_ + t) * G4_ + 4 * hcol);
        const float Gi = s4.x + x4.x, Gf = s4.y + x4.y;
        const float Gg = s4.z + x4.z, Go = s4.w + x4.w;
        dacc[j][p] += Deltas[((size_t)b * T_ + t) * H_ + hcol];
        const float a     = atj[b * T_ + t];
        const float alpha = a / logf(2.7182818284590452f + dacc[j][p]);
        float c = alpha * c0v[j][p] + (1.f - alpha) * cv[j][p];
        c = sigm(Gf) * c + sigm(Gi) * tanhf(Gg);
        const float h = sigm(Go) * tanhf(c);
        cv[j][p]   = c;
        hreg[j][p] = h;
        hn[((size_t)b * T_ + t) * H_ + hcol] = h;
      }
    }
    __syncthreads();   // all hbuf reads of step t done
#pragma unroll
    for (int j = 0; j < 2; ++j)
#pragma unroll
      for (int p = 0; p < 8; ++p) {
        const int b    = rt[j] * 16 + pr[p];
        const int hcol = cg[j] * 16 + ph[p];
        hbuf[b * HP + hcol] = hreg[j][p];
      }
    __syncthreads();   // hbuf writes visible before next step's reads
  }
}

// ---------------------------------------------------------------------------
// Launcher
// ---------------------------------------------------------------------------
extern "C" void kernel_launch(void* const* d_in, const int* in_sizes, int n_in,
                              void* d_out, int out_size, void* d_ws,
                              size_t ws_size, hipStream_t stream) {
  const float* values = (const float*)d_in[0];
  const float* Deltas = (const float*)d_in[1];
  const float* W_att  = (const float*)d_in[2];
  const float* b_att  = (const float*)d_in[3];
  const float* W_ih   = (const float*)d_in[4];
  const float* W_hh   = (const float*)d_in[5];
  const float* b_ih   = (const float*)d_in[6];
  const float* b_hh   = (const float*)d_in[7];
  float* hn = (float*)d_out;

  const int BT = B_ * T_;
  float* ws   = (float*)d_ws;
  float* vlin = ws;                 size_t o = (size_t)BT * D_;
  float* Xg   = ws + o;             o += (size_t)BT * G4_;
  float* atj  = ws + o;             o += BT;
  float* WpI  = ws + o;             o += (size_t)G4_ * D_;
  float* WpH  = ws + o;             o += (size_t)G4_ * H_;
  float* bp   = ws + o;

  pack_kernel<<<dim3(G4_), dim3(D_), 0, stream>>>(W_ih, W_hh, b_ih, b_hh, WpI,
                                                  WpH, bp);
  gemm_bias_kernel<<<dim3(D_ / 64, BT / 64), dim3(128), 0, stream>>>(
      values, W_att, b_att, vlin, D_, D_);
  attn_kernel<<<dim3(B_), dim3(256), 0, stream>>>(vlin, values, atj);
  gemm_bias_kernel<<<dim3(G4_ / 64, BT / 64), dim3(128), 0, stream>>>(
      vlin, WpI, bp, Xg, D_, G4_);
  const size_t ldsBytes = (size_t)(64 * HP + 32 * 16 * SP) * sizeof(float);
  (void)hipFuncSetAttribute((const void*)scan_kernel,
                            hipFuncAttributeMaxDynamicSharedMemorySize,
                            (int)ldsBytes);
  scan_kernel<<<dim3(1), dim3(1024), ldsBytes, stream>>>(Xg, WpH, atj, Deltas,
                                                         hn);
}